// VectorQuantizer_3813930959166
// MI455X (gfx1250) — compile-verified
//
#include <hip/hip_runtime.h>

typedef float v2f __attribute__((ext_vector_type(2)));
typedef float v8f __attribute__((ext_vector_type(8)));

#define NUM_EMB 1024
#define EMB_DIM 64
#define N_ROWS (16 * 128 * 128)          // 262144 flattened input rows
#define Q_ELEMS (N_ROWS * EMB_DIM)       // 16777216 quantized elements
#define ROWS_PER_WAVE 32                 // two 16-row A tiles share each B tile
#define WAVES_PER_BLOCK 8
#define ROWS_PER_BLOCK (ROWS_PER_WAVE * WAVES_PER_BLOCK)   // 256
#define NUM_BLOCKS (N_ROWS / ROWS_PER_BLOCK)               // 1024
#define NUM_WAVES_TOTAL (NUM_BLOCKS * WAVES_PER_BLOCK)     // 8192
#define COMMIT_SCALE 1.25f               // q_latent + 0.25 * e_latent (same forward value)

// ---------------------------------------------------------------------------
// Kernel 1: per-code squared norms  WS[k] = ||W[k]||^2   (K=1024, D=64)
// ---------------------------------------------------------------------------
__global__ void vq_wsum_kernel(const float* __restrict__ W, float* __restrict__ WS) {
    int k = blockIdx.x * blockDim.x + threadIdx.x;
    if (k < NUM_EMB) {
        const float* wr = W + k * EMB_DIM;
        float s = 0.0f;
#pragma unroll
        for (int d = 0; d < EMB_DIM; ++d) s += wr[d] * wr[d];
        WS[k] = s;
    }
}

// ---------------------------------------------------------------------------
// Kernel 2: main WMMA distance + argmin + gather + loss partials.
// One wave = 32 rows (two 16x64 A tiles).  Loops 64 tiles of 16 codes;
// each B tile feeds 32 V_WMMA_F32_16X16X4_F32 (two independent 16-deep
// accumulation chains -- C-chaining is hazard-free on CDNA5).
// ---------------------------------------------------------------------------
__global__ __launch_bounds__(256) void vq_main_kernel(
        const float* __restrict__ X, const float* __restrict__ W,
        const float* __restrict__ WS, float* __restrict__ Qout,
        float* __restrict__ IdxOut, float* __restrict__ partial) {
    const int lane = threadIdx.x & 31;
    const int wave = threadIdx.x >> 5;
    const int lh   = lane & 15;   // column within half
    const int hi   = lane >> 4;   // which half of the wave
    const int rowbase = (blockIdx.x * WAVES_PER_BLOCK + wave) * ROWS_PER_WAVE;

    // --- Preload A fragments for both 16-row tiles (16x4 f32 layout:
    // lane l (0-15): row = l, K={0,1}; lane l+16: same row, K={2,3}).
    const float* xrow0 = X + (size_t)(rowbase + lh) * EMB_DIM + 2 * hi;
    const float* xrow1 = xrow0 + 16 * EMB_DIM;
    v2f a0[16], a1[16];
#pragma unroll
    for (int c = 0; c < 16; ++c) {
        a0[c] = *(const v2f*)(xrow0 + 4 * c);
        a1[c] = *(const v2f*)(xrow1 + 4 * c);
    }

    float runmin0[8], runmin1[8];
    int   runidx0[8], runidx1[8];
#pragma unroll
    for (int r = 0; r < 8; ++r) {
        runmin0[r] = 3.4e38f; runidx0[r] = 0;
        runmin1[r] = 3.4e38f; runidx1[r] = 0;
    }

    for (int j = 0; j < 64; ++j) {
        const int codebase = j * 16;
        const float* wrow = W + (size_t)(codebase + lh) * EMB_DIM + 2 * hi;
        if (j + 1 < 64)
            __builtin_prefetch(wrow + 16 * EMB_DIM, 0, 1);  // next B tile

        // B fragments: B[k][n] = W[codebase+n][4c+k], mirrored 4x16 layout.
        v2f b[16];
#pragma unroll
        for (int c = 0; c < 16; ++c)
            b[c] = *(const v2f*)(wrow + 4 * c);

        v8f acc0 = {};
        v8f acc1 = {};
#pragma unroll
        for (int c = 0; c < 16; ++c) {
            acc0 = __builtin_amdgcn_wmma_f32_16x16x4_f32(
                false, a0[c], false, b[c], (short)0, acc0, false, false);
            acc1 = __builtin_amdgcn_wmma_f32_16x16x4_f32(
                false, a1[c], false, b[c], (short)0, acc1, false, false);
        }

        // distance (dropping row-constant ||x||^2): ||e||^2 - 2 x.e
        const int   code = codebase + lh;
        const float wv   = WS[code];
#pragma unroll
        for (int r = 0; r < 8; ++r) {
            float d0 = __builtin_fmaf(-2.0f, acc0[r], wv);
            float d1 = __builtin_fmaf(-2.0f, acc1[r], wv);
            if (d0 < runmin0[r]) { runmin0[r] = d0; runidx0[r] = code; }
            if (d1 < runmin1[r]) { runmin1[r] = d1; runidx1[r] = code; }
        }
    }

    // --- Cross-lane argmin within each 16-lane half, then gather winning
    // code rows, write quantized + indices, and accumulate the loss partial.
    float sq = 0.0f;
#pragma unroll
    for (int t = 0; t < 2; ++t) {
#pragma unroll
        for (int r = 0; r < 8; ++r) {
            float v = t ? runmin1[r] : runmin0[r];
            int   i = t ? runidx1[r] : runidx0[r];
#pragma unroll
            for (int m = 1; m <= 8; m <<= 1) {
                float ov = __shfl_xor(v, m, 32);
                int   oi = __shfl_xor(i, m, 32);
                if (ov < v || (ov == v && oi < i)) { v = ov; i = oi; }
            }
            const int row = rowbase + 16 * t + 8 * hi + r;
            const float* wq = W + (size_t)i * EMB_DIM;
            const float* xr = X + (size_t)row * EMB_DIM;
            float*       qo = Qout + (size_t)row * EMB_DIM;
#pragma unroll
            for (int u = 0; u < 4; ++u) {
                const int dcol = lh + 16 * u;
                const float q  = wq[dcol];
                const float df = q - xr[dcol];
                qo[dcol] = q;
                sq = __builtin_fmaf(df, df, sq);
            }
            if (lh == 0) IdxOut[row] = (float)i;
        }
    }
#pragma unroll
    for (int m = 1; m < 32; m <<= 1) sq += __shfl_xor(sq, m, 32);
    if (lane == 0) partial[blockIdx.x * WAVES_PER_BLOCK + wave] = sq;
}

// ---------------------------------------------------------------------------
// Kernel 3: deterministic fold of 8192 per-wave partials -> scalar loss.
// ---------------------------------------------------------------------------
__global__ void vq_loss_kernel(const float* __restrict__ partial,
                               float* __restrict__ loss_out) {
    const int lane = threadIdx.x;            // 32 threads
    const int per  = NUM_WAVES_TOTAL / 32;   // 256 contiguous per lane
    const float* p = partial + lane * per;
    float s = 0.0f;
    for (int i = 0; i < per; ++i) s += p[i];
#pragma unroll
    for (int m = 1; m < 32; m <<= 1) s += __shfl_xor(s, m, 32);
    if (lane == 0) *loss_out = COMMIT_SCALE * (s / (float)Q_ELEMS);
}

// ---------------------------------------------------------------------------
extern "C" void kernel_launch(void* const* d_in, const int* in_sizes, int n_in,
                              void* d_out, int out_size, void* d_ws, size_t ws_size,
                              hipStream_t stream) {
    const float* X = (const float*)d_in[0];   // [262144, 64]
    const float* W = (const float*)d_in[1];   // [1024, 64]

    float* out      = (float*)d_out;
    float* Qout     = out;                    // 16777216 floats
    float* loss_out = out + Q_ELEMS;          // 1 float
    float* IdxOut   = out + Q_ELEMS + 1;      // 262144 indices (as float values)

    float* WS      = (float*)d_ws;            // 1024 floats: ||e_k||^2
    float* partial = WS + NUM_EMB;            // 8192 floats: per-wave loss partials

    vq_wsum_kernel<<<(NUM_EMB + 255) / 256, 256, 0, stream>>>(W, WS);
    vq_main_kernel<<<NUM_BLOCKS, 256, 0, stream>>>(X, W, WS, Qout, IdxOut, partial);
    vq_loss_kernel<<<1, 32, 0, stream>>>(partial, loss_out);
}